// MMD_Alignment_7954279432505
// MI455X (gfx1250) — compile-verified
//
#include <hip/hip_runtime.h>
#include <hip/hip_bf16.h>

typedef float v2f __attribute__((ext_vector_type(2)));
typedef float v8f __attribute__((ext_vector_type(8)));

#define NS    4096
#define NTOT  8192
#define DIMK  256

// ---------------------------------------------------------------------------
// Kernel 1: sq[i] = sum_k T[i,k]^2   (T = concat(source, target), row-major)
// One 256-thread block per row; LDS tree reduction.
// ---------------------------------------------------------------------------
__global__ void mmd_sq_kernel(const float* __restrict__ src,
                              const float* __restrict__ tgt,
                              float* __restrict__ sq) {
    __shared__ float red[256];
    const int row = blockIdx.x;
    const float* p = (row < NS) ? (src + (size_t)row * DIMK)
                                : (tgt + (size_t)(row - NS) * DIMK);
    float x = p[threadIdx.x];
    red[threadIdx.x] = x * x;
    __syncthreads();
    for (int s = 128; s > 0; s >>= 1) {
        if (threadIdx.x < s) red[threadIdx.x] += red[threadIdx.x + s];
        __syncthreads();
    }
    if (threadIdx.x == 0) sq[row] = red[0];
}

// ---------------------------------------------------------------------------
// Gram pass via V_WMMA_F32_16X16X4_F32.
// Block = 256 threads = 8 waves in a 2x4 grid; each wave owns a 2x2 set of
// 16x16 f32 accumulators -> block tile is 64 (rows) x 128 (cols).
//
// f32 16x4 A-fragment layout (ISA 7.12.2): lane L, elem v holds
//   A[M = L%16, K = 2*(L/16) + v]
// f32 4x16 B-fragment layout: lane L, elem v holds
//   B[K = 2*(L/16) + v, N = L%16]
// For Gram (B[k,n] = T[j0+n, k]) both fragments therefore load the SAME
// pattern from row-major T: float2 at T[(base + L%16)*256 + k0 + 2*(L/16)].
//
// PASS 0: accumulate sum(max(L2,0))            -> accum[0] (f64 atomics)
// PASS 1: accumulate signed 5-kernel exp sum   -> accum[1] (f64 atomics)
// ---------------------------------------------------------------------------
template <int PASS>
__global__ void mmd_gram_kernel(const float* __restrict__ src,
                                const float* __restrict__ tgt,
                                const float* __restrict__ sq,
                                double* __restrict__ accum) {
    const int lane = threadIdx.x & 31;
    const int w    = threadIdx.x >> 5;
    const int wm   = w & 1;        // 0..1
    const int wn   = w >> 1;       // 0..3
    const int i0   = blockIdx.y * 64  + wm * 32;
    const int j0   = blockIdx.x * 128 + wn * 32;
    const int r    = lane & 15;
    const int half = lane >> 4;

    // Per-lane row base pointers (each 16-row fragment group lies entirely
    // in source or target since 4096 % 16 == 0, but per-lane select is fine).
    const float* arow[2];
    const float* brow[2];
#pragma unroll
    for (int t = 0; t < 2; ++t) {
        int ri = i0 + t * 16 + r;
        arow[t] = ((ri < NS) ? (src + (size_t)ri * DIMK)
                             : (tgt + (size_t)(ri - NS) * DIMK)) + 2 * half;
        int rj = j0 + t * 16 + r;
        brow[t] = ((rj < NS) ? (src + (size_t)rj * DIMK)
                             : (tgt + (size_t)(rj - NS) * DIMK)) + 2 * half;
    }

    v8f acc[2][2] = {};
    for (int k0 = 0; k0 < DIMK; k0 += 4) {
        v2f a0 = *(const v2f*)(arow[0] + k0);
        v2f a1 = *(const v2f*)(arow[1] + k0);
        v2f b0 = *(const v2f*)(brow[0] + k0);
        v2f b1 = *(const v2f*)(brow[1] + k0);
        acc[0][0] = __builtin_amdgcn_wmma_f32_16x16x4_f32(
            false, a0, false, b0, (short)0, acc[0][0], false, false);
        acc[0][1] = __builtin_amdgcn_wmma_f32_16x16x4_f32(
            false, a0, false, b1, (short)0, acc[0][1], false, false);
        acc[1][0] = __builtin_amdgcn_wmma_f32_16x16x4_f32(
            false, a1, false, b0, (short)0, acc[1][0], false, false);
        acc[1][1] = __builtin_amdgcn_wmma_f32_16x16x4_f32(
            false, a1, false, b1, (short)0, acc[1][1], false, false);
    }

    float inv16 = 0.0f;
    if (PASS == 1) {
        // bw0 = sum(L2) / (N^2 - N) / KERNEL_MUL^(KERNEL_NUM/2)
        double bw0 = accum[0] / ((double)NTOT * (double)NTOT - (double)NTOT) / 4.0;
        inv16 = (float)(1.0 / (bw0 * 16.0));   // widest kernel's 1/denominator
    }

    float part = 0.0f;
#pragma unroll
    for (int mi = 0; mi < 2; ++mi) {
#pragma unroll
        for (int ni = 0; ni < 2; ++ni) {
            const int mbase = i0 + mi * 16 + 8 * half;   // C/D layout: M = e + 8*half
            const int n     = j0 + ni * 16 + r;          //             N = lane%16
            const float sqn = sq[n];
#pragma unroll
            for (int e = 0; e < 8; ++e) {
                const int m  = mbase + e;
                float g  = acc[mi][ni][e];
                float l2 = sq[m] + sqn - 2.0f * g;
                l2 = l2 > 0.0f ? l2 : 0.0f;
                if (PASS == 0) {
                    part += l2;
                } else {
                    // sum_{i=0..4} exp(-l2/(bw0*2^i)) = t + t^2 + t^4 + t^8 + t^16
                    // with t = exp(-l2/(16*bw0)); one v_exp_f32 + 4 squarings.
                    float t = __expf(-l2 * inv16);
                    float u = t, s5 = t;
                    u *= u; s5 += u;   // t^2
                    u *= u; s5 += u;   // t^4
                    u *= u; s5 += u;   // t^8
                    u *= u; s5 += u;   // t^16
                    bool same = (m < NS) == (n < NS);    // XX/YY: +, XY/YX: -
                    part += same ? s5 : -s5;
                }
            }
        }
    }

    // wave32 reduction, then one f64 global atomic per wave
    for (int off = 16; off > 0; off >>= 1)
        part += __shfl_down(part, off, 32);
    if (lane == 0)
        atomicAdd(&accum[(PASS == 0) ? 0 : 1], (double)part);
}

// ---------------------------------------------------------------------------
// Kernel 4: out = signedSum / ns^2
// ---------------------------------------------------------------------------
__global__ void mmd_finalize_kernel(const double* __restrict__ accum,
                                    float* __restrict__ out) {
    if (threadIdx.x == 0 && blockIdx.x == 0)
        out[0] = (float)(accum[1] / ((double)NS * (double)NS));
}

extern "C" void kernel_launch(void* const* d_in, const int* in_sizes, int n_in,
                              void* d_out, int out_size, void* d_ws, size_t ws_size,
                              hipStream_t stream) {
    const float* src = (const float*)d_in[0];
    const float* tgt = (const float*)d_in[1];
    float* out = (float*)d_out;

    char*   ws    = (char*)d_ws;
    float*  sq    = (float*)ws;                    // 8192 floats = 32 KB
    double* accum = (double*)(ws + 8192 * sizeof(float)); // [0]=sumL2 [1]=signed

    hipMemsetAsync(accum, 0, 2 * sizeof(double), stream);

    mmd_sq_kernel<<<NTOT, 256, 0, stream>>>(src, tgt, sq);

    dim3 grid(NTOT / 128, NTOT / 64);   // 64 x 128
    mmd_gram_kernel<0><<<grid, 256, 0, stream>>>(src, tgt, sq, accum);
    mmd_gram_kernel<1><<<grid, 256, 0, stream>>>(src, tgt, sq, accum);

    mmd_finalize_kernel<<<1, 32, 0, stream>>>(accum, out);
}